// HNMS_60224031424605
// MI455X (gfx1250) — compile-verified
//
#include <hip/hip_runtime.h>
#include <stdint.h>

// ---------------------------------------------------------------------------
// HNMS: hash boxes into quantized cells, keep per-cell argmax(conf) with
// min-index tie-break. Lock-free hash table in d_ws sized to stay resident in
// MI455X's 192MB L2; winner reduction via global_atomic_max_u64 on a monotone
// (conf_bits << 32 | ~idx) packing. Streaming input/output traffic uses
// non-temporal cache hints (gfx1250 TH=NT) so it does not evict the table.
// ---------------------------------------------------------------------------

#define HNMS_BLOCK 256  // 8 wave32 waves per block

typedef float f32x4 __attribute__((ext_vector_type(4)));

__device__ __forceinline__ unsigned long long hnms_pack16(float v) {
    long long t = (long long)v + 32768ll;          // _OFF
    t = t < 0 ? 0 : t;
    t = t > 65535 ? 65535 : t;
    return (unsigned long long)t;
}

// Quantize (cx,cy,w,h) -> 64-bit cell key.
//   i_w = rint(log(w)/log(a)) = rint(log2(w) * 1/log2(a))
//   1/cell_w = a^(-i_w) = exp2(-i_w * log2(a))
//   i_x = rint(x / (0.5*cell_w)) = rint(x * 2 * inv_cell_w)
// v_log_f32 (= log2) and v_exp_f32 (= exp2) via amdgcn builtins:
// 2 logs + 2 exps + a few FMAs; no divisions, no libm calls.
__device__ __forceinline__ unsigned long long hnms_key(f32x4 r) {
    const float kInvL2A = -1.9433582038804653f;    // 1 / log2(0.7)
    const float kNegL2A =  0.5145731728297583f;    // -log2(0.7)
    float i_w = rintf(__builtin_amdgcn_logf(r.z) * kInvL2A);   // v_log_f32
    float i_h = rintf(__builtin_amdgcn_logf(r.w) * kInvL2A);
    float i_x = rintf(r.x * 2.0f * __builtin_amdgcn_exp2f(i_w * kNegL2A)); // v_exp_f32
    float i_y = rintf(r.y * 2.0f * __builtin_amdgcn_exp2f(i_h * kNegL2A));
    return (hnms_pack16(i_w) << 48) | (hnms_pack16(i_h) << 32) |
           (hnms_pack16(i_x) << 16) |  hnms_pack16(i_y);
}

// MurmurHash3 fmix64 finalizer: full-avalanche hash of the 64-bit key.
__device__ __forceinline__ unsigned long long hnms_mix(unsigned long long h) {
    h ^= h >> 33; h *= 0xff51afd7ed558ccdULL;
    h ^= h >> 33; h *= 0xc4ceb9fe1a85ec53ULL;
    h ^= h >> 33;
    return h;
}

__global__ __launch_bounds__(HNMS_BLOCK)
void hnms_init_table(unsigned long long* __restrict__ keys,
                     unsigned long long* __restrict__ vals,
                     unsigned int cap) {
    unsigned int i = blockIdx.x * HNMS_BLOCK + threadIdx.x;
    if (i < cap) {
        keys[i] = 0ULL;    // 0 is unreachable as a real key (fields ~0x8000)
        vals[i] = 0ULL;    // below any real packed value (idx < 2^32-1)
    }
}

// Pass 1: claim a slot per unique key (linear probe + 64-bit CAS), then reduce
// the packed (conf_bits<<32 | ~idx) value with a 64-bit atomic max.
// Commutative reduction -> deterministic table contents regardless of races.
__global__ __launch_bounds__(HNMS_BLOCK)
void hnms_insert(const f32x4* __restrict__ rects,
                 const float* __restrict__ conf,
                 unsigned long long* __restrict__ keys,
                 unsigned long long* __restrict__ vals,
                 unsigned int mask, unsigned int n) {
    unsigned int i = blockIdx.x * HNMS_BLOCK + threadIdx.x;
    if (i >= n) return;

    f32x4 r = __builtin_nontemporal_load(rects + i);   // b128, TH=NT
    float  c = __builtin_nontemporal_load(conf + i);   // TH=NT
    unsigned long long key = hnms_key(r);
    unsigned long long packed =
        ((unsigned long long)__float_as_uint(c) << 32) |
        (unsigned long long)(~i);

    unsigned int slot = (unsigned int)hnms_mix(key) & mask;
    for (unsigned int probes = 0; probes <= mask; ++probes) {
        unsigned long long prev = atomicCAS(&keys[slot], 0ULL, key);
        if (prev == 0ULL || prev == key) {
            atomicMax(&vals[slot], packed);            // global_atomic_max_u64
            return;
        }
        slot = (slot + 1) & mask;
    }
    // Table overflow (unreachable with cap >= 2x distinct keys).
}

// Pass 2: re-derive key, probe the now read-only L2-resident table, and emit
// keep = (cell winner == me).
__global__ __launch_bounds__(HNMS_BLOCK)
void hnms_resolve(const f32x4* __restrict__ rects,
                  const float* __restrict__ conf,
                  const unsigned long long* __restrict__ keys,
                  const unsigned long long* __restrict__ vals,
                  float* __restrict__ out,
                  unsigned int mask, unsigned int n) {
    unsigned int i = blockIdx.x * HNMS_BLOCK + threadIdx.x;
    if (i >= n) return;

    f32x4 r = __builtin_nontemporal_load(rects + i);   // TH=NT
    float  c = __builtin_nontemporal_load(conf + i);   // TH=NT
    unsigned long long key = hnms_key(r);
    unsigned long long packed =
        ((unsigned long long)__float_as_uint(c) << 32) |
        (unsigned long long)(~i);

    unsigned int slot = (unsigned int)hnms_mix(key) & mask;
    float keep = 0.0f;
    for (unsigned int probes = 0; probes <= mask; ++probes) {
        unsigned long long k = keys[slot];
        if (k == key) {
            keep = (vals[slot] == packed) ? 1.0f : 0.0f;
            break;
        }
        if (k == 0ULL) break;                          // not found (unreachable)
        slot = (slot + 1) & mask;
    }
    __builtin_nontemporal_store(keep, out + i);        // TH=NT store
}

extern "C" void kernel_launch(void* const* d_in, const int* in_sizes, int n_in,
                              void* d_out, int out_size, void* d_ws, size_t ws_size,
                              hipStream_t stream) {
    const f32x4* rects = (const f32x4*)d_in[0];   // (N,4) f32, 16B records
    const float* conf  = (const float*)d_in[1];   // (N,)  f32
    unsigned int n = (unsigned int)in_sizes[1];

    // Hash table capacity: largest power of two fitting in d_ws, capped at
    // 8M slots (128MB -> fully resident in MI455X's 192MB L2).
    size_t cap = (size_t)1 << 23;
    while (cap * 16 > ws_size && cap > 4096) cap >>= 1;
    unsigned int mask = (unsigned int)cap - 1u;

    unsigned long long* keys = (unsigned long long*)d_ws;
    unsigned long long* vals = keys + cap;

    dim3 block(HNMS_BLOCK);
    dim3 grid_tab((unsigned int)((cap + HNMS_BLOCK - 1) / HNMS_BLOCK));
    dim3 grid_n((n + HNMS_BLOCK - 1) / HNMS_BLOCK);

    hnms_init_table<<<grid_tab, block, 0, stream>>>(keys, vals, (unsigned int)cap);
    hnms_insert   <<<grid_n,   block, 0, stream>>>(rects, conf, keys, vals, mask, n);
    hnms_resolve  <<<grid_n,   block, 0, stream>>>(rects, conf, keys, vals,
                                                   (float*)d_out, mask, n);
}